// KGEFuseBlock_46127948759326
// MI455X (gfx1250) — compile-verified
//
#include <hip/hip_runtime.h>
#include <math.h>

typedef __attribute__((ext_vector_type(2))) float v2f;
typedef __attribute__((ext_vector_type(8))) float v8f;

// ---------------------------------------------------------------------------
// Inner K-loop for the 32x32-tile-per-wave WMMA GEMM.  GUARD=false is the
// interior-tile fast path: no masking at all, 8 independent global loads per
// 4 WMMAs.  GUARD=true zero-fills out-of-range lanes via select (loads are
// unconditional from clamped addresses, so EXEC stays all-ones for WMMA).
// Distinct temporaries per load keep all 8 loads in flight simultaneously.
// ---------------------------------------------------------------------------
template <bool GUARD>
__device__ __forceinline__ void gemm_kloop(const float*& pA0, const float*& pA1,
                                           const float*& pB0, const float*& pB1,
                                           long lsAk, long lsBk, int Kmain,
                                           bool okA0, bool okA1, bool okB0, bool okB1,
                                           v8f& acc00, v8f& acc01,
                                           v8f& acc10, v8f& acc11) {
#pragma unroll 2
  for (int k0 = 0; k0 < Kmain; k0 += 4) {
    const float tA0x = pA0[0], tA0y = pA0[lsAk];
    const float tA1x = pA1[0], tA1y = pA1[lsAk];
    const float tB0x = pB0[0], tB0y = pB0[lsBk];
    const float tB1x = pB1[0], tB1y = pB1[lsBk];
    v2f a0, a1, b0, b1;
    if (GUARD) {
      a0.x = okA0 ? tA0x : 0.f;  a0.y = okA0 ? tA0y : 0.f;
      a1.x = okA1 ? tA1x : 0.f;  a1.y = okA1 ? tA1y : 0.f;
      b0.x = okB0 ? tB0x : 0.f;  b0.y = okB0 ? tB0y : 0.f;
      b1.x = okB1 ? tB1x : 0.f;  b1.y = okB1 ? tB1y : 0.f;
    } else {
      a0.x = tA0x; a0.y = tA0y;
      a1.x = tA1x; a1.y = tA1y;
      b0.x = tB0x; b0.y = tB0y;
      b1.x = tB1x; b1.y = tB1y;
    }
    acc00 = __builtin_amdgcn_wmma_f32_16x16x4_f32(false, a0, false, b0, (short)0, acc00, false, false);
    acc01 = __builtin_amdgcn_wmma_f32_16x16x4_f32(false, a0, false, b1, (short)0, acc01, false, false);
    acc10 = __builtin_amdgcn_wmma_f32_16x16x4_f32(false, a1, false, b0, (short)0, acc10, false, false);
    acc11 = __builtin_amdgcn_wmma_f32_16x16x4_f32(false, a1, false, b1, (short)0, acc11, false, false);
    pA0 += 4 * lsAk; pA1 += 4 * lsAk;
    pB0 += 4 * lsBk; pB1 += 4 * lsBk;
  }
}

// ---------------------------------------------------------------------------
// Generic strided f32 WMMA GEMM: C[b,m,n] = sum_k A[b,m,k] * B[b,k,n]
// One wave per 32x32 C-tile, V_WMMA_F32_16X16X4_F32, 4 accumulators.
// Interior tiles take the unmasked fast loop (wave-uniform branch).
// ---------------------------------------------------------------------------
__global__ void wmma_gemm_f32_kernel(const float* __restrict__ A,
                                     const float* __restrict__ B,
                                     float* __restrict__ C,
                                     int M, int N, int Kd,
                                     int sAm, int sAk,
                                     int sBk, int sBn,
                                     int sCm, int sCn,
                                     long sAb, long sBb, long sCb) {
  const int lane = threadIdx.x & 31;
  const int wave = blockIdx.x * (blockDim.x >> 5) + (threadIdx.x >> 5);
  const int tilesN = (N + 31) >> 5;
  const int tilesM = (M + 31) >> 5;
  if (wave >= tilesM * tilesN) return;   // wave-uniform: EXEC stays all-ones

  const int tm = (wave / tilesN) << 5;
  const int tn = (wave % tilesN) << 5;

  const float* Ab = A + (long)blockIdx.y * sAb;
  const float* Bb = B + (long)blockIdx.y * sBb;
  float*       Cb = C + (long)blockIdx.y * sCb;

  // A 16x4 layout: lanes 0-15 hold (row=lane, K=0/1), lanes 16-31 (row, K=2/3)
  // B 4x16 layout: lanes 0-15 hold (K=0/1, col=lane), lanes 16-31 (K=2/3, col)
  const int half  = lane >> 4;
  const int idx16 = lane & 15;
  const int kb    = half * 2;

  const int r0 = tm + idx16;
  const int r1 = tm + 16 + idx16;
  const int c0 = tn + idx16;
  const int c1 = tn + 16 + idx16;
  const bool okA0 = r0 < M, okA1 = r1 < M;
  const bool okB0 = c0 < N, okB1 = c1 < N;
  const int r0c = okA0 ? r0 : (M - 1);
  const int r1c = okA1 ? r1 : (M - 1);
  const int c0c = okB0 ? c0 : (N - 1);
  const int c1c = okB1 ? c1 : (N - 1);

  const long lsAk = sAk, lsBk = sBk;
  const float* pA0 = Ab + (long)r0c * sAm + (long)kb * lsAk;
  const float* pA1 = Ab + (long)r1c * sAm + (long)kb * lsAk;
  const float* pB0 = Bb + (long)c0c * sBn + (long)kb * lsBk;
  const float* pB1 = Bb + (long)c1c * sBn + (long)kb * lsBk;

  v8f acc00 = {0.f,0.f,0.f,0.f,0.f,0.f,0.f,0.f};
  v8f acc01 = acc00, acc10 = acc00, acc11 = acc00;

  const int Kmain = Kd & ~3;
  const bool interior = (tm + 32 <= M) && (tn + 32 <= N);  // wave-uniform
  if (interior) {
    gemm_kloop<false>(pA0, pA1, pB0, pB1, lsAk, lsBk, Kmain,
                      okA0, okA1, okB0, okB1, acc00, acc01, acc10, acc11);
  } else {
    gemm_kloop<true>(pA0, pA1, pB0, pB1, lsAk, lsBk, Kmain,
                     okA0, okA1, okB0, okB1, acc00, acc01, acc10, acc11);
  }

  if (Kmain < Kd) {                    // masked K-tail (Kd % 4 != 0)
    const int kt0 = Kmain + kb;
    const int kt1 = kt0 + 1;
    const bool k0ok = kt0 < Kd, k1ok = kt1 < Kd;
    const long dk0 = (long)((k0ok ? kt0 : Kd - 1) - kt0);  // 0 or negative delta
    const long dk1 = (long)((k1ok ? kt1 : Kd - 1) - kt0);
    const float tA00 = pA0[dk0 * lsAk], tA01 = pA0[dk1 * lsAk];
    const float tA10 = pA1[dk0 * lsAk], tA11 = pA1[dk1 * lsAk];
    const float tB00 = pB0[dk0 * lsBk], tB01 = pB0[dk1 * lsBk];
    const float tB10 = pB1[dk0 * lsBk], tB11 = pB1[dk1 * lsBk];
    v2f a0, a1, b0, b1;
    a0.x = (okA0 && k0ok) ? tA00 : 0.f;  a0.y = (okA0 && k1ok) ? tA01 : 0.f;
    a1.x = (okA1 && k0ok) ? tA10 : 0.f;  a1.y = (okA1 && k1ok) ? tA11 : 0.f;
    b0.x = (okB0 && k0ok) ? tB00 : 0.f;  b0.y = (okB0 && k1ok) ? tB01 : 0.f;
    b1.x = (okB1 && k0ok) ? tB10 : 0.f;  b1.y = (okB1 && k1ok) ? tB11 : 0.f;
    acc00 = __builtin_amdgcn_wmma_f32_16x16x4_f32(false, a0, false, b0, (short)0, acc00, false, false);
    acc01 = __builtin_amdgcn_wmma_f32_16x16x4_f32(false, a0, false, b1, (short)0, acc01, false, false);
    acc10 = __builtin_amdgcn_wmma_f32_16x16x4_f32(false, a1, false, b0, (short)0, acc10, false, false);
    acc11 = __builtin_amdgcn_wmma_f32_16x16x4_f32(false, a1, false, b1, (short)0, acc11, false, false);
  }

  // C/D layout: VGPR v -> row (v) for lanes 0-15, row (v+8) for lanes 16-31
  const int rb0 = tm + half * 8;
  const int rb1 = tm + 16 + half * 8;
#pragma unroll
  for (int v = 0; v < 8; ++v) {
    const int ra = rb0 + v;
    const int rc = rb1 + v;
    if (ra < M && okB0) Cb[(long)ra * sCm + (long)c0 * sCn] = acc00[v];
    if (ra < M && okB1) Cb[(long)ra * sCm + (long)c1 * sCn] = acc01[v];
    if (rc < M && okB0) Cb[(long)rc * sCm + (long)c0 * sCn] = acc10[v];
    if (rc < M && okB1) Cb[(long)rc * sCm + (long)c1 * sCn] = acc11[v];
  }
}

// ---------------------------------------------------------------------------
// Depthwise causal conv (kernel 4) + bias + SiLU.
// ---------------------------------------------------------------------------
__global__ void conv_silu_kernel(const float* __restrict__ xz,
                                 const float* __restrict__ w,
                                 const float* __restrict__ bias,
                                 float* __restrict__ out,
                                 int Bn, int L, int C) {
  long idx = (long)blockIdx.x * blockDim.x + threadIdx.x;
  const long total = (long)Bn * L * C;
  if (idx >= total) return;
  const int  c  = (int)(idx % C);
  const long bl = idx / C;
  const int  l  = (int)(bl % L);
  const long b  = bl / L;
  float acc = bias[c];
#pragma unroll
  for (int j = 0; j < 4; ++j) {
    const int ls = l - 3 + j;
    if (ls >= 0) acc += xz[(b * L + ls) * (long)(2 * C) + c] * w[c * 4 + j];
  }
  out[idx] = acc * (1.f / (1.f + __expf(-acc)));   // SiLU
}

// dt = softplus(dt_pre + bias), in place
__global__ void softplus_bias_kernel(float* __restrict__ dt,
                                     const float* __restrict__ bias,
                                     long M, int C) {
  long idx = (long)blockIdx.x * blockDim.x + threadIdx.x;
  if (idx >= M * C) return;
  const int c = (int)(idx % C);
  const float x = dt[idx] + bias[c];
  dt[idx] = (x > 20.f) ? x : log1pf(__expf(x));
}

// ---------------------------------------------------------------------------
// Selective scan: one thread per (batch, channel); 16 states in registers.
// ---------------------------------------------------------------------------
__global__ void scan_kernel(const float* __restrict__ u,
                            const float* __restrict__ dt,
                            const float* __restrict__ xdbl, int X, int boff, int coff,
                            const float* __restrict__ A_log,
                            const float* __restrict__ Dv,
                            float* __restrict__ y,
                            int Bn, int L, int C) {
  long idx = (long)blockIdx.x * blockDim.x + threadIdx.x;
  if (idx >= (long)Bn * C) return;
  const int c = (int)(idx % C);
  const int b = (int)(idx / C);

  float Ar[16];
#pragma unroll
  for (int n = 0; n < 16; ++n) Ar[n] = -__expf(A_log[c * 16 + n]);
  const float Dd = Dv[c];

  float h[16];
#pragma unroll
  for (int n = 0; n < 16; ++n) h[n] = 0.f;

  for (int l = 0; l < L; ++l) {
    const long base = (long)b * L + l;
    const float dtv = dt[base * C + c];
    const float uu  = u[base * C + c];
    const float* xd = xdbl + base * X;
    float yv = 0.f;
#pragma unroll
    for (int n = 0; n < 16; ++n) {
      const float dA = __expf(dtv * Ar[n]);
      h[n] = dA * h[n] + dtv * xd[boff + n] * uu;
      yv += h[n] * xd[coff + n];
    }
    y[base * C + c] = yv + uu * Dd;
  }
}

// g = y * silu(z), z = second half of xz
__global__ void gate_kernel(const float* __restrict__ y,
                            const float* __restrict__ xz,
                            float* __restrict__ g,
                            long M, int C) {
  long idx = (long)blockIdx.x * blockDim.x + threadIdx.x;
  if (idx >= M * C) return;
  const int  c = (int)(idx % C);
  const long m = idx / C;
  const float z = xz[m * (long)(2 * C) + C + c];
  g[idx] = y[idx] * z * (1.f / (1.f + __expf(-z)));
}

// mean over last axis (N) then softmax over K, one block per batch (K threads)
__global__ void mean_softmax_kernel(const float* __restrict__ score,
                                    float* __restrict__ s, int K, int N) {
  __shared__ float red[1024];
  const int b = blockIdx.x;
  const int t = threadIdx.x;
  const float* row = score + ((long)b * K + t) * N;
  float m = 0.f;
  for (int n = 0; n < N; ++n) m += row[n];
  m /= (float)N;
  red[t] = m; __syncthreads();
  for (int st = 512; st > 0; st >>= 1) {
    if (t < st) red[t] = fmaxf(red[t], red[t + st]);
    __syncthreads();
  }
  const float mx = red[0]; __syncthreads();
  const float e = __expf(m - mx);
  red[t] = e; __syncthreads();
  for (int st = 512; st > 0; st >>= 1) {
    if (t < st) red[t] += red[t + st];
    __syncthreads();
  }
  s[(long)b * K + t] = e / red[0];
}

// iterative top-R (descending, lowest index wins ties), one block per batch
__global__ void topk_kernel(const float* __restrict__ s,
                            float* __restrict__ vals, int* __restrict__ inds,
                            int K, int R) {
  __shared__ float sv[1024];
  __shared__ float rv[256];
  __shared__ int   ri[256];
  const int b = blockIdx.x;
  const int t = threadIdx.x;
  for (int i = t; i < K; i += 256) sv[i] = s[(long)b * K + i];
  __syncthreads();
  for (int r = 0; r < R; ++r) {
    float bv = -INFINITY; int bi = 0;
    for (int i = t; i < K; i += 256) {
      const float v = sv[i];
      if (v > bv) { bv = v; bi = i; }
    }
    rv[t] = bv; ri[t] = bi; __syncthreads();
    for (int st = 128; st > 0; st >>= 1) {
      if (t < st) {
        if (rv[t + st] > rv[t] ||
            (rv[t + st] == rv[t] && ri[t + st] < ri[t])) {
          rv[t] = rv[t + st]; ri[t] = ri[t + st];
        }
      }
      __syncthreads();
    }
    if (t == 0) {
      vals[b * R + r] = rv[0];
      inds[b * R + r] = ri[0];
      sv[ri[0]] = -INFINITY;
    }
    __syncthreads();
  }
}

// h[b,i,:] = imageFeature[i,:] + kf[b,:,ind[b,i]] * vtop[b,i]
__global__ void gather_h_kernel(const float* __restrict__ IF,
                                const float* __restrict__ kf,
                                const float* __restrict__ vals,
                                const int* __restrict__ inds,
                                float* __restrict__ h,
                                int Bn, int Npos, int DM, int K) {
  long idx = (long)blockIdx.x * blockDim.x + threadIdx.x;
  const long total = (long)Bn * Npos * DM;
  if (idx >= total) return;
  const int d = (int)(idx % DM);
  const long rest = idx / DM;
  const int i = (int)(rest % Npos);
  const int b = (int)(rest / Npos);
  const int k = inds[b * Npos + i];
  const float v = vals[b * Npos + i];
  h[idx] = IF[(long)i * DM + d] + kf[((long)b * DM + d) * K + k] * v;
}

// ---------------------------------------------------------------------------
// Host-side orchestration
// ---------------------------------------------------------------------------
static inline void launch_gemm(hipStream_t stream,
                               const float* A, const float* B, float* C,
                               int M, int N, int Kd,
                               int sAm, int sAk, int sBk, int sBn,
                               int sCm, int sCn,
                               long sAb, long sBb, long sCb, int batches) {
  const int tiles = ((M + 31) / 32) * ((N + 31) / 32);
  const int wavesPerBlock = 8;                 // 256 threads = 8 wave32
  const int blocks = (tiles + wavesPerBlock - 1) / wavesPerBlock;
  dim3 grid(blocks, batches);
  wmma_gemm_f32_kernel<<<grid, dim3(256), 0, stream>>>(
      A, B, C, M, N, Kd, sAm, sAk, sBk, sBn, sCm, sCn, sAb, sBb, sCb);
}

static inline int eblocks(long total) { return (int)((total + 255) / 256); }

extern "C" void kernel_launch(void* const* d_in, const int* in_sizes, int n_in,
                              void* d_out, int out_size, void* d_ws, size_t ws_size,
                              hipStream_t stream) {
  (void)in_sizes; (void)n_in; (void)out_size; (void)ws_size;

  const float* IF       = (const float*)d_in[0];   // (98, 2048)
  const float* kf       = (const float*)d_in[1];   // (8, 2048, 1024)
  const float* a_in_w   = (const float*)d_in[2];   // (98, 392)
  const float* a_conv_w = (const float*)d_in[3];   // (196, 4)
  const float* a_conv_b = (const float*)d_in[4];   // (196,)
  const float* a_xproj  = (const float*)d_in[5];   // (196, 39)
  const float* a_dt_w   = (const float*)d_in[6];   // (7, 196)
  const float* a_dt_b   = (const float*)d_in[7];   // (196,)
  const float* a_A_log  = (const float*)d_in[8];   // (196, 16)
  const float* a_D      = (const float*)d_in[9];   // (196,)
  const float* a_out_w  = (const float*)d_in[10];  // (196, 98)
  const float* c_in_w   = (const float*)d_in[11];  // (2048, 8192)
  const float* c_conv_w = (const float*)d_in[12];  // (4096, 4)
  const float* c_conv_b = (const float*)d_in[13];  // (4096,)
  const float* c_xproj  = (const float*)d_in[14];  // (4096, 160)
  const float* c_dt_w   = (const float*)d_in[15];  // (128, 4096)
  const float* c_dt_b   = (const float*)d_in[16];  // (4096,)
  const float* c_A_log  = (const float*)d_in[17];  // (4096, 16)
  const float* c_D      = (const float*)d_in[18];  // (4096,)
  const float* c_out_w  = (const float*)d_in[19];  // (4096, 2048)

  float* out = (float*)d_out;                      // (8, 98, 2048)

  const int B = 8, K = 1024, N = 98, DM = 2048;
  const int aC = 196, aR = 7;
  const int cC = 4096, cR = 128;
  const long Ma = (long)B * K;                     // 8192
  const long Mc = (long)B * N;                     // 784

  float* ws = (float*)d_ws;
  long off = 0;
  float* score  = ws + off; off += (long)B * K * N;
  float* xz_a   = ws + off; off += Ma * 2 * aC;
  float* xi_a   = ws + off; off += Ma * aC;
  float* xdbl_a = ws + off; off += Ma * (aR + 32);
  float* dt_a   = ws + off; off += Ma * aC;
  float* y_a    = ws + off; off += Ma * aC;
  float* g_a    = ws + off; off += Ma * aC;
  float* score2 = ws + off; off += (long)B * K * N;
  float* svec   = ws + off; off += (long)B * K;
  float* topv   = ws + off; off += (long)B * N;
  int*   topi   = (int*)(ws + off); off += (long)B * N;
  float* hbuf   = ws + off; off += Mc * DM;
  float* xz_c   = ws + off; off += Mc * 2 * cC;
  float* xi_c   = ws + off; off += Mc * cC;
  float* xdbl_c = ws + off; off += Mc * (cR + 32);
  float* dt_c   = ws + off; off += Mc * cC;
  float* y_c    = ws + off; off += Mc * cC;

  // 1) score[b,k,n] = sum_d kf[b,d,k] * IF[n,d]
  launch_gemm(stream, kf, IF, score, K, N, DM,
              1, K, 1, DM, N, 1, (long)DM * K, 0, (long)K * N, B);

  // ---- Mamba A ----
  launch_gemm(stream, score, a_in_w, xz_a, (int)Ma, 2 * aC, N,
              N, 1, 2 * aC, 1, 2 * aC, 1, 0, 0, 0, 1);
  conv_silu_kernel<<<eblocks(Ma * aC), 256, 0, stream>>>(xz_a, a_conv_w, a_conv_b,
                                                         xi_a, B, K, aC);
  launch_gemm(stream, xi_a, a_xproj, xdbl_a, (int)Ma, aR + 32, aC,
              aC, 1, aR + 32, 1, aR + 32, 1, 0, 0, 0, 1);
  launch_gemm(stream, xdbl_a, a_dt_w, dt_a, (int)Ma, aC, aR,
              aR + 32, 1, aC, 1, aC, 1, 0, 0, 0, 1);
  softplus_bias_kernel<<<eblocks(Ma * aC), 256, 0, stream>>>(dt_a, a_dt_b, Ma, aC);
  scan_kernel<<<eblocks((long)B * aC), 256, 0, stream>>>(
      xi_a, dt_a, xdbl_a, aR + 32, aR, aR + 16, a_A_log, a_D, y_a, B, K, aC);
  gate_kernel<<<eblocks(Ma * aC), 256, 0, stream>>>(y_a, xz_a, g_a, Ma, aC);
  launch_gemm(stream, g_a, a_out_w, score2, (int)Ma, N, aC,
              aC, 1, N, 1, N, 1, 0, 0, 0, 1);

  // ---- selection ----
  mean_softmax_kernel<<<B, 1024, 0, stream>>>(score2, svec, K, N);
  topk_kernel<<<B, 256, 0, stream>>>(svec, topv, topi, K, N);
  gather_h_kernel<<<eblocks(Mc * DM), 256, 0, stream>>>(IF, kf, topv, topi,
                                                        hbuf, B, N, DM, K);

  // ---- Mamba C ----
  launch_gemm(stream, hbuf, c_in_w, xz_c, (int)Mc, 2 * cC, DM,
              DM, 1, 2 * cC, 1, 2 * cC, 1, 0, 0, 0, 1);
  conv_silu_kernel<<<eblocks(Mc * cC), 256, 0, stream>>>(xz_c, c_conv_w, c_conv_b,
                                                         xi_c, B, N, cC);
  launch_gemm(stream, xi_c, c_xproj, xdbl_c, (int)Mc, cR + 32, cC,
              cC, 1, cR + 32, 1, cR + 32, 1, 0, 0, 0, 1);
  launch_gemm(stream, xdbl_c, c_dt_w, dt_c, (int)Mc, cC, cR,
              cR + 32, 1, cC, 1, cC, 1, 0, 0, 0, 1);
  softplus_bias_kernel<<<eblocks(Mc * cC), 256, 0, stream>>>(dt_c, c_dt_b, Mc, cC);
  scan_kernel<<<eblocks((long)B * cC), 256, 0, stream>>>(
      xi_c, dt_c, xdbl_c, cR + 32, cR, cR + 16, c_A_log, c_D, y_c, B, N, cC);
  gate_kernel<<<eblocks(Mc * cC), 256, 0, stream>>>(y_c, xz_c, xi_c, Mc, cC);
  launch_gemm(stream, xi_c, c_out_w, out, (int)Mc, DM, cC,
              cC, 1, DM, 1, DM, 1, 0, 0, 0, 1);
}